// Deep_Pron_45681272161029
// MI455X (gfx1250) — compile-verified
//
#include <hip/hip_runtime.h>
#include <cmath>

#define N_  32
#define D_  1128
#define V_  100
#define NF_ 13
#define H_  1000
#define EPS_ 1e-5f

typedef __attribute__((ext_vector_type(2))) float v2f;
typedef __attribute__((ext_vector_type(4))) float v4f;
typedef __attribute__((ext_vector_type(8))) float v8f;
typedef int ai4 __attribute__((vector_size(16)));   // matches builtin param type

// ---------------------------------------------------------------------------
// CDNA5 async global->LDS staging (ASYNCcnt path), with safe fallback.
// ---------------------------------------------------------------------------
#if defined(__has_builtin)
#if __has_builtin(__builtin_amdgcn_global_load_async_to_lds_b128) && \
    __has_builtin(__builtin_amdgcn_s_wait_asynccnt)
#define USE_ASYNC_LDS 1
#endif
#endif

// Copies nfloats (multiple of 4, 16B-aligned both sides) from global to LDS.
// Each wave waits for its own ASYNCcnt; caller must __syncthreads() after.
__device__ __forceinline__ void lds_stage(float* __restrict__ dst,
                                          const float* __restrict__ src,
                                          int nfloats, int tid, int nthreads) {
#ifdef USE_ASYNC_LDS
    typedef __attribute__((address_space(1))) ai4* gp_t;
    typedef __attribute__((address_space(3))) ai4* lp_t;
    gp_t g = (gp_t)const_cast<float*>(src);
    lp_t l = (lp_t)dst;
    const int nchunks = nfloats >> 2;
    for (int i = tid; i < nchunks; i += nthreads)
        __builtin_amdgcn_global_load_async_to_lds_b128(g + i, l + i, 0, 0);
    __builtin_amdgcn_s_wait_asynccnt(0);
#else
    const int nchunks = nfloats >> 2;
    const v4f* g4 = (const v4f*)src;
    v4f* d4 = (v4f*)dst;
    for (int i = tid; i < nchunks; i += nthreads) d4[i] = g4[i];
#endif
}

// ---------------------------------------------------------------------------
// Kernel 1 (fused): per-channel BN2d stats THEN attention for all 32 batch
// rows of that channel.  grid = D_, block = 256.  The channel's 332KB slice is
// read from HBM once (stats) and re-read L2-hot (attend) -> ~1 HBM pass total.
// Note x^T A x == x^T A^T x, so attn_w is used un-transposed.
// ---------------------------------------------------------------------------
__global__ void channel_kernel(const float* __restrict__ X1,
                               const float* __restrict__ X2,
                               const float* __restrict__ M1,
                               const float* __restrict__ M2,
                               const float* __restrict__ Aw,
                               const float* __restrict__ gamma,
                               const float* __restrict__ beta,
                               float* __restrict__ feats) {
    const int d = blockIdx.x;
    const int t = threadIdx.x;          // 256 threads = 8 wave32
    __shared__ __align__(16) float xs1[V_ * NF_];
    __shared__ __align__(16) float xs2[V_ * NF_];
    __shared__ float Am[NF_ * NF_];
    __shared__ float r1[256], r2[256];
    __shared__ float prm[4];            // al1, be1, al2, be2

    const float inv = 1.0f / (float)(N_ * V_ * NF_);

    // ---- phase 1: channel stats for X1 and X2 ----
    for (int which = 0; which < 2; ++which) {
        const float* X = which ? X2 : X1;
        float s = 0.f, q = 0.f;
        for (int n = 0; n < N_; ++n) {
            const float* base = X + ((size_t)n * D_ + d) * (size_t)(V_ * NF_);
            for (int i = t; i < V_ * NF_; i += 256) {
                float x = base[i];
                s += x; q += x * x;
            }
        }
        r1[t] = s; r2[t] = q;
        __syncthreads();
        for (int off = 128; off > 0; off >>= 1) {
            if (t < off) { r1[t] += r1[t + off]; r2[t] += r2[t + off]; }
            __syncthreads();
        }
        if (t == 0) {
            float m   = r1[0] * inv;
            float var = r2[0] * inv - m * m;
            float al  = gamma[d] * rsqrtf(var + EPS_);
            prm[2 * which]     = al;
            prm[2 * which + 1] = beta[d] - m * al;
        }
        __syncthreads();
    }
    const float al1 = prm[0], be1 = prm[1], al2 = prm[2], be2 = prm[3];
    for (int i = t; i < NF_ * NF_; i += 256) Am[i] = Aw[i];

    // ---- phase 2: attention per batch row (slices now L2-hot) ----
    for (int n = 0; n < N_; ++n) {
        const size_t base = ((size_t)n * D_ + d) * (size_t)(V_ * NF_);
        __syncthreads();                         // previous iter done with xs
        lds_stage(xs1, X1 + base, V_ * NF_, t, 256);
        lds_stage(xs2, X2 + base, V_ * NF_, t, 256);
        __syncthreads();
        for (int i = t; i < V_ * NF_; i += 256) {   // normalize in place
            xs1[i] = xs1[i] * al1 + be1;
            xs2[i] = xs2[i] * al2 + be2;
        }
        __syncthreads();

        float lg1 = -3.0e38f, lg2 = -3.0e38f;
        if (t < V_) {
            const float* x1 = &xs1[t * NF_];
            const float* x2 = &xs2[t * NF_];
            float s1 = 0.f, s2 = 0.f;
            #pragma unroll
            for (int i = 0; i < NF_; ++i) {
                float ax1 = 0.f, ax2 = 0.f;
                #pragma unroll
                for (int j = 0; j < NF_; ++j) {
                    float a = Am[i * NF_ + j];
                    ax1 = fmaf(a, x1[j], ax1);
                    ax2 = fmaf(a, x2[j], ax2);
                }
                s1 = fmaf(x1[i], ax1, s1);
                s2 = fmaf(x2[i], ax2, s2);
            }
            // mask touches only feature 0 -> strided scalar loads
            float m1 = M1[base + (size_t)t * NF_];
            float m2 = M2[base + (size_t)t * NF_];
            lg1 = tanhf(s1) + (m1 - 1.0f) * 100000.0f;
            lg2 = tanhf(s2) + (m2 - 1.0f) * 100000.0f;
        }
        r1[t] = lg1; r2[t] = lg2;
        __syncthreads();
        for (int off = 128; off > 0; off >>= 1) {
            if (t < off) { r1[t] = fmaxf(r1[t], r1[t+off]); r2[t] = fmaxf(r2[t], r2[t+off]); }
            __syncthreads();
        }
        const float mx1 = r1[0], mx2 = r2[0];
        __syncthreads();
        float e1 = (t < V_) ? expf(lg1 - mx1) : 0.f;
        float e2 = (t < V_) ? expf(lg2 - mx2) : 0.f;
        r1[t] = e1; r2[t] = e2;
        __syncthreads();
        for (int off = 128; off > 0; off >>= 1) {
            if (t < off) { r1[t] += r1[t+off]; r2[t] += r2[t+off]; }
            __syncthreads();
        }
        const float sum1 = r1[0], sum2 = r2[0];
        __syncthreads();
        if (t < V_) {
            const float w1 = e1 / sum1;
            const float w2 = e2 / sum2;
            #pragma unroll
            for (int i = 0; i < NF_; ++i) {
                xs1[t * NF_ + i] *= w1;
                xs2[t * NF_ + i] *= w2;
            }
        }
        __syncthreads();
        if (t < NF_) {
            float h1 = 0.f, h2 = 0.f;
            for (int v = 0; v < V_; ++v) {
                h1 += xs1[v * NF_ + t];
                h2 += xs2[v * NF_ + t];
            }
            float dd = h1 - h2;
            r1[t] = dd * dd;
        }
        __syncthreads();
        if (t == 0) {
            float ss = 0.f;
            #pragma unroll
            for (int i = 0; i < NF_; ++i) ss += r1[i];
            float f  = logf(ss + 1e-5f);
            float pm = (M1[base] + M2[base] == 2.0f) ? 1.0f : 0.0f;
            feats[(size_t)n * D_ + d] = (f + 1.0f) * pm - 1.0f;
        }
    }
}

// ---------------------------------------------------------------------------
// Kernel 2: BatchNorm1d over the batch (N=32) per feature d.
// ---------------------------------------------------------------------------
__global__ void bn1d_kernel(const float* __restrict__ feats,
                            const float* __restrict__ gamma,
                            const float* __restrict__ beta,
                            float* __restrict__ xbn) {
    const int d = blockIdx.x * blockDim.x + threadIdx.x;
    if (d >= D_) return;
    float s = 0.f;
    for (int n = 0; n < N_; ++n) s += feats[n * D_ + d];
    const float m = s * (1.0f / N_);
    float q = 0.f;
    for (int n = 0; n < N_; ++n) {
        float x = feats[n * D_ + d] - m;
        q += x * x;
    }
    const float al = gamma[d] * rsqrtf(q * (1.0f / N_) + EPS_);
    const float be = beta[d];
    for (int n = 0; n < N_; ++n)
        xbn[n * D_ + d] = (feats[n * D_ + d] - m) * al + be;
}

// ---------------------------------------------------------------------------
// Kernel 3: out[32,Hout] = act(in[32,K] @ W^T + bias) via V_WMMA_F32_16X16X4_F32.
// Activations async-staged into LDS (<=144KB of CDNA5's 320KB WGP LDS).
// A frag (16x4 f32): lanes 0-15 -> K=k,k+1 ; lanes 16-31 -> K=k+2,k+3.
// B frag mirrors on the N side.  C/D: VGPR r = row (r, r+8), col = lane&15.
// block = 256 (8 wave32), grid = 16.
// ---------------------------------------------------------------------------
__global__ void gemm_wmma_kernel(const float* __restrict__ in,
                                 const float* __restrict__ W,
                                 const float* __restrict__ bias,
                                 float* __restrict__ out,
                                 int K, int Hout, int do_relu) {
    extern __shared__ __align__(16) float s_in[];
    const int tid = threadIdx.x;
    lds_stage(s_in, in, N_ * K, tid, blockDim.x);
    __syncthreads();

    const int wave = tid >> 5;
    const int lane = tid & 31;
    const int half = lane >> 4;       // 0: K pair (k,k+1), 1: (k+2,k+3)
    const int lr   = lane & 15;

    const int nTiles = (Hout + 15) >> 4;
    const int totalTiles = 2 * nTiles;        // 2 M-tiles (M = 32)
    const int gw = blockIdx.x * (blockDim.x >> 5) + wave;
    const int nw = gridDim.x * (blockDim.x >> 5);

    for (int tile = gw; tile < totalTiles; tile += nw) {
        const int mt = tile & 1;
        const int nt = tile >> 1;
        const int mrow = mt * 16 + lr;                 // A row for this lane
        const int ncol = nt * 16 + lr;                 // output column
        const int ncl  = (ncol < Hout) ? ncol : (Hout - 1);  // clamp pad lanes
        const float* wrow = W + (size_t)ncl * (size_t)K;     // row of W = col of W^T
        const float* arow = s_in + (size_t)mrow * (size_t)K;
        __builtin_prefetch(wrow, 0, 1);                // global_prefetch_b8

        v8f c = {0.f, 0.f, 0.f, 0.f, 0.f, 0.f, 0.f, 0.f};
        for (int k = 0; k < K; k += 4) {
            const int kk = k + (half << 1);
            v2f a  = *(const v2f*)(arow + kk);         // ds_load_b64
            v2f bb = *(const v2f*)(wrow + kk);         // global_load_b64
            c = __builtin_amdgcn_wmma_f32_16x16x4_f32(
                    false, a, false, bb, (short)0, c, false, false);
        }
        if (ncol < Hout) {
            const float bn = bias[ncol];
            #pragma unroll
            for (int r = 0; r < 8; ++r) {
                const int m = mt * 16 + r + half * 8;
                float v = c[r] + bn;
                if (do_relu) v = fmaxf(v, 0.f);
                out[(size_t)m * (size_t)Hout + ncol] = v;
            }
        }
    }
}

// ---------------------------------------------------------------------------
// Kernel 4: final fc7 (Hout=1) dot product per batch row.
// ---------------------------------------------------------------------------
__global__ void fc7_kernel(const float* __restrict__ h,
                           const float* __restrict__ w,
                           const float* __restrict__ b,
                           float* __restrict__ out) {
    const int n = blockIdx.x;
    const int t = threadIdx.x;   // 128
    __shared__ float red[128];
    float s = 0.f;
    for (int k = t; k < H_; k += 128) s = fmaf(h[n * H_ + k], w[k], s);
    red[t] = s;
    __syncthreads();
    for (int off = 64; off > 0; off >>= 1) {
        if (t < off) red[t] += red[t + off];
        __syncthreads();
    }
    if (t == 0) out[n] = red[0] + b[0];
}

extern "C" void kernel_launch(void* const* d_in, const int* in_sizes, int n_in,
                              void* d_out, int out_size, void* d_ws, size_t ws_size,
                              hipStream_t stream) {
    const float* X1     = (const float*)d_in[0];
    const float* X2     = (const float*)d_in[1];
    const float* M1     = (const float*)d_in[2];
    const float* M2     = (const float*)d_in[3];
    const float* attn_w = (const float*)d_in[4];
    const float* g2d    = (const float*)d_in[5];
    const float* b2d    = (const float*)d_in[6];
    const float* g1d    = (const float*)d_in[7];
    const float* b1d    = (const float*)d_in[8];
    const float* fc1_w  = (const float*)d_in[9];
    const float* fc1_b  = (const float*)d_in[10];
    const float* fc2_w  = (const float*)d_in[11];
    const float* fc2_b  = (const float*)d_in[12];
    const float* fc3_w  = (const float*)d_in[13];
    const float* fc3_b  = (const float*)d_in[14];
    const float* fc4_w  = (const float*)d_in[15];
    const float* fc4_b  = (const float*)d_in[16];
    const float* fc5_w  = (const float*)d_in[17];
    const float* fc5_b  = (const float*)d_in[18];
    const float* fc6_w  = (const float*)d_in[19];
    const float* fc6_b  = (const float*)d_in[20];
    const float* fc7_w  = (const float*)d_in[21];
    const float* fc7_b  = (const float*)d_in[22];
    float* out = (float*)d_out;

    float* ws    = (float*)d_ws;
    float* feats = ws;                          // [N, D]
    float* xbn   = feats + (size_t)N_ * D_;     // [N, D]
    float* hA    = xbn   + (size_t)N_ * D_;     // [N, H]
    float* hB    = hA    + (size_t)N_ * H_;     // [N, H]

    channel_kernel<<<D_, 256, 0, stream>>>(X1, X2, M1, M2, attn_w, g2d, b2d, feats);
    bn1d_kernel<<<(D_ + 127) / 128, 128, 0, stream>>>(feats, g1d, b1d, xbn);

    const size_t shm1 = (size_t)N_ * D_ * sizeof(float);   // 144 KB
    const size_t shmH = (size_t)N_ * H_ * sizeof(float);   // 128 KB
    gemm_wmma_kernel<<<16, 256, shm1, stream>>>(xbn, fc1_w, fc1_b, hA, D_, H_, 1);
    gemm_wmma_kernel<<<16, 256, shmH, stream>>>(hA, fc2_w, fc2_b, hB, H_, H_, 1);
    gemm_wmma_kernel<<<16, 256, shmH, stream>>>(hB, fc3_w, fc3_b, hA, H_, H_, 1);
    gemm_wmma_kernel<<<16, 256, shmH, stream>>>(hA, fc4_w, fc4_b, hB, H_, H_, 1);
    gemm_wmma_kernel<<<16, 256, shmH, stream>>>(hB, fc5_w, fc5_b, hA, H_, H_, 1);
    gemm_wmma_kernel<<<16, 256, shmH, stream>>>(hA, fc6_w, fc6_b, hB, H_, H_, 1);
    fc7_kernel<<<N_, 128, 0, stream>>>(hB, fc7_w, fc7_b, out);
}